// HiPPO_9646496547006
// MI455X (gfx1250) — compile-verified
//
#include <hip/hip_runtime.h>

// HiPPO-LegT scan: c_t = dA c_{t-1} + dB x_t, emit all c_t.
// L=8192, BS=256, N=64. Output 512MB f32 -> memory floor ~22us @23.3TB/s.
// fp32 WMMA (16x16x4) is mandatory: dA = I + O(1e-4), bf16 would drop the
// recurrence dynamics entirely. Chunked parallel scan:
//   k_prep: P = dA^64 (6 WMMA squarings, 1 WG)
//   k_end : per (chunk,btile) local scan w/ zero init -> chunk-end states
//   k_scan: prefix over 128 chunk boundaries via P (1 WG, 16 indep waves)
//   k_emit: replay chunks from correct start states, stream outputs.
// State tile is state-major 64x16 (M=state, N=batch) = 4 WMMA C-tiles/wave.
// Feeding C-layout output back as next step's B operand costs exactly one
// ds_swizzle SWAPX16 + one v_cndmask per B VGPR (32+32 ops per 64 WMMAs).

typedef __attribute__((ext_vector_type(2))) float v2f;
typedef __attribute__((ext_vector_type(4))) float v4f;
typedef __attribute__((ext_vector_type(8))) float v8f;

#define L_LEN 8192
#define BS_   256
#define NST   64
#define TCH   64            // chunk length
#define GCH   128           // L_LEN / TCH
#define BT    16            // BS_ / 16 batch tiles

// ds_swizzle SWAPX16: group-of-32, and=0x1f, or=0, xor=0x10 -> 0x7c10
__device__ __forceinline__ float swapx16(float v) {
  return __int_as_float(__builtin_amdgcn_ds_swizzle(__float_as_int(v), 0x7c10));
}

__device__ __forceinline__ v8f wmma4(v2f a, v2f b, v8f c) {
  // D = A(16x4 f32) x B(4x16 f32) + C(16x16 f32)
  return __builtin_amdgcn_wmma_f32_16x16x4_f32(false, a, false, b, (short)0, c,
                                               false, false);
}

// Build B fragment (rows 4k..4k+3 of the 64-row state, all 16 batch cols)
// from 4 C-layout tiles. Within a 16x16 C tile: row a<8 -> VGPR a lanes0-15,
// row a>=8 -> VGPR a-8 lanes16-31. B VGPR0 wants row a in lanes0-15 and row
// a+1 in lanes16-31 (VGPR1: rows a+2,a+3).
__device__ __forceinline__ v2f make_bfrag(const v8f cin[4], int k, bool hi) {
  const int tt = k >> 2;        // which 16-row tile
  const int q  = k & 3;         // row quad within tile
  const int r0 = (q & 1) ? 4 : 0;
  const v8f t = cin[tt];
  v2f b;
  if (q < 2) {                  // rows live in lanes0-15 of VGPRs r0..r0+3
    float s1 = swapx16(t[r0 + 1]);
    float s3 = swapx16(t[r0 + 3]);
    b.x = hi ? s1 : t[r0 + 0];
    b.y = hi ? s3 : t[r0 + 2];
  } else {                      // rows live in lanes16-31 of VGPRs r0..r0+3
    float s0 = swapx16(t[r0 + 0]);
    float s2 = swapx16(t[r0 + 2]);
    b.x = hi ? t[r0 + 1] : s0;
    b.y = hi ? t[r0 + 3] : s2;
  }
  return b;
}

// A fragments of a 64x64 row-major matrix M: tile i covers output rows
// 16i..16i+15, frag k covers K = 4k..4k+3.  A layout: lane l%16 = row,
// VGPR0 holds K = 4k + (l<16 ? 0 : 2), VGPR1 the next K.
__device__ __forceinline__ void load_afrags(const float* __restrict__ M,
                                            v2f A[4][16], int lane) {
  const int ml = lane & 15;
  const int koff = (lane & 16) ? 2 : 0;
#pragma unroll
  for (int i = 0; i < 4; ++i)
#pragma unroll
    for (int k = 0; k < 16; ++k)
      A[i][k] = *(const v2f*)(M + (size_t)(16 * i + ml) * NST + 4 * k + koff);
}

// dB replicated into C-layout: component r of tile i holds dB[16i+hi8+r].
__device__ __forceinline__ void load_dbv(const float* __restrict__ dB,
                                         v8f dbv[4], int lane) {
  const int hb = (lane & 16) ? 8 : 0;
#pragma unroll
  for (int i = 0; i < 4; ++i) {
    v4f lo = *(const v4f*)(dB + 16 * i + hb);
    v4f hq = *(const v4f*)(dB + 16 * i + hb + 4);
    dbv[i] = __builtin_shufflevector(lo, hq, 0, 1, 2, 3, 4, 5, 6, 7);
  }
}

// acc[i] += (A-matrix) @ cin ;  64 x v_wmma_f32_16x16x4_f32
__device__ __forceinline__ void mat_step(const v2f A[4][16], const v8f cin[4],
                                         v8f acc[4], bool hi) {
#pragma unroll
  for (int k = 0; k < 16; ++k) {
    v2f b = make_bfrag(cin, k, hi);
#pragma unroll
    for (int i = 0; i < 4; ++i) acc[i] = wmma4(A[i][k], b, acc[i]);
  }
}

// ---------------- prep: P = dA^64 via 6 WMMA squarings (1 WG, 4 waves) -----
__global__ void __launch_bounds__(128) hippo_prep(const float* __restrict__ dA,
                                                  float* __restrict__ P) {
  __shared__ float buf0[4096];
  __shared__ float buf1[4096];
  const int tid = threadIdx.x;
  const int lane = tid & 31, wv = tid >> 5;
  const bool hi = (lane & 16) != 0;
  const int ml = lane & 15;
  for (int j = tid; j < 4096; j += 128) buf0[j] = dA[j];
  __syncthreads();
#pragma unroll 1
  for (int it = 0; it < 6; ++it) {
    const float* M = (it & 1) ? buf1 : buf0;
    float* D = (it & 1) ? buf0 : buf1;
    const int koff = hi ? 2 : 0;
    v8f acc[4];
    const v8f z = {0, 0, 0, 0, 0, 0, 0, 0};
#pragma unroll
    for (int i = 0; i < 4; ++i) acc[i] = z;
#pragma unroll
    for (int k = 0; k < 16; ++k) {
      v2f b;  // B layout: VGPR0 lane: (K=4k + hi, N=16*wv+ml), VGPR1: K+2
      b.x = M[(4 * k + (hi ? 1 : 0)) * NST + 16 * wv + ml];
      b.y = M[(4 * k + 2 + (hi ? 1 : 0)) * NST + 16 * wv + ml];
#pragma unroll
      for (int i = 0; i < 4; ++i) {
        v2f a = *(const v2f*)(M + (16 * i + ml) * NST + 4 * k + koff);
        acc[i] = wmma4(a, b, acc[i]);
      }
    }
    const int hb = hi ? 8 : 0;
#pragma unroll
    for (int i = 0; i < 4; ++i)
#pragma unroll
      for (int r = 0; r < 8; ++r)
        D[(16 * i + hb + r) * NST + 16 * wv + ml] = acc[i][r];
    __syncthreads();
  }
  for (int j = tid; j < 4096; j += 128) P[j] = buf0[j];  // it=5 wrote buf0
}

// ------ k1: chunk-local scan, zero init, store only end state to S --------
// S blob per (chunk,btile) wid: 1024 floats, per-lane 32 contiguous floats
// flat = lane*32 + i*8 + r  -> two b128 per tile.
__global__ void __launch_bounds__(32) hippo_chunk_end(
    const float* __restrict__ X, const float* __restrict__ dA,
    const float* __restrict__ dB, float* __restrict__ S) {
  const int wid = blockIdx.x;
  const int g = wid >> 4, bt = wid & (BT - 1);
  const int lane = threadIdx.x;
  const bool hi = (lane & 16) != 0;
  v2f A[4][16];
  load_afrags(dA, A, lane);
  v8f dbv[4];
  load_dbv(dB, dbv, lane);
  v8f c[4];
  const v8f z = {0, 0, 0, 0, 0, 0, 0, 0};
#pragma unroll
  for (int i = 0; i < 4; ++i) c[i] = z;
  const float* xp = X + (size_t)g * TCH * BS_ + bt * 16 + (lane & 15);
#pragma unroll 1
  for (int t = 0; t < TCH; ++t) {
    __builtin_prefetch(xp + (size_t)(t + 6) * BS_, 0, 1);
    const float xv = xp[(size_t)t * BS_];
    v8f acc[4];
#pragma unroll
    for (int i = 0; i < 4; ++i) acc[i] = dbv[i] * xv;  // u_t = dB * x_t
    mat_step(A, c, acc, hi);
#pragma unroll
    for (int i = 0; i < 4; ++i) c[i] = acc[i];
  }
  float* sp = S + (size_t)wid * 1024 + lane * 32;
#pragma unroll
  for (int i = 0; i < 4; ++i) {
    *(v4f*)(sp + i * 8)     = __builtin_shufflevector(c[i], c[i], 0, 1, 2, 3);
    *(v4f*)(sp + i * 8 + 4) = __builtin_shufflevector(c[i], c[i], 4, 5, 6, 7);
  }
}

// ------ k2: prefix over chunk boundaries: I[g] = P@I[g-1] + end[g] --------
// In-place: S[g] gets start state I[g-1].  16 independent waves (one/btile).
__global__ void __launch_bounds__(512) hippo_scan(const float* __restrict__ P,
                                                  float* __restrict__ S) {
  const int lane = threadIdx.x & 31;
  const int bt = threadIdx.x >> 5;
  const bool hi = (lane & 16) != 0;
  v2f A[4][16];
  load_afrags(P, A, lane);
  v8f I[4];
  const v8f z = {0, 0, 0, 0, 0, 0, 0, 0};
#pragma unroll
  for (int i = 0; i < 4; ++i) I[i] = z;
#pragma unroll 1
  for (int g = 0; g < GCH; ++g) {
    float* sp = S + (size_t)(g * BT + bt) * 1024 + lane * 32;
    v8f e[4];
#pragma unroll
    for (int i = 0; i < 4; ++i) {
      v4f lo = *(const v4f*)(sp + i * 8);
      v4f hq = *(const v4f*)(sp + i * 8 + 4);
      e[i] = __builtin_shufflevector(lo, hq, 0, 1, 2, 3, 4, 5, 6, 7);
    }
    // load->store to same address: only store->load ordering is guaranteed
    // by HW, so drain LOADcnt before overwriting with the start state.
    asm volatile("s_wait_loadcnt 0x0" ::: "memory");
#pragma unroll
    for (int i = 0; i < 4; ++i) {
      *(v4f*)(sp + i * 8)     = __builtin_shufflevector(I[i], I[i], 0, 1, 2, 3);
      *(v4f*)(sp + i * 8 + 4) = __builtin_shufflevector(I[i], I[i], 4, 5, 6, 7);
    }
    v8f acc[4];
#pragma unroll
    for (int i = 0; i < 4; ++i) acc[i] = e[i];
    mat_step(A, I, acc, hi);
#pragma unroll
    for (int i = 0; i < 4; ++i) I[i] = acc[i];
  }
}

// ------ k3: replay chunk from correct start state, stream all outputs -----
__global__ void __launch_bounds__(32) hippo_emit(
    const float* __restrict__ X, const float* __restrict__ dA,
    const float* __restrict__ dB, const float* __restrict__ S,
    float* __restrict__ OUT) {
  const int wid = blockIdx.x;
  const int g = wid >> 4, bt = wid & (BT - 1);
  const int lane = threadIdx.x;
  const bool hi = (lane & 16) != 0;
  const int hb = hi ? 8 : 0;
  v2f A[4][16];
  load_afrags(dA, A, lane);
  v8f dbv[4];
  load_dbv(dB, dbv, lane);
  v8f c[4];
  const float* sp = S + (size_t)wid * 1024 + lane * 32;
#pragma unroll
  for (int i = 0; i < 4; ++i) {
    v4f lo = *(const v4f*)(sp + i * 8);
    v4f hq = *(const v4f*)(sp + i * 8 + 4);
    c[i] = __builtin_shufflevector(lo, hq, 0, 1, 2, 3, 4, 5, 6, 7);
  }
  const int col = bt * 16 + (lane & 15);
  const float* xp = X + (size_t)g * TCH * BS_ + col;
  float* ob = OUT + ((size_t)g * TCH * BS_ + col) * NST;
#pragma unroll 1
  for (int t = 0; t < TCH; ++t) {
    __builtin_prefetch(xp + (size_t)(t + 6) * BS_, 0, 1);
    const float xv = xp[(size_t)t * BS_];
    v8f acc[4];
#pragma unroll
    for (int i = 0; i < 4; ++i) acc[i] = dbv[i] * xv;
    mat_step(A, c, acc, hi);
#pragma unroll
    for (int i = 0; i < 4; ++i) c[i] = acc[i];
    // per lane: tile i covers states n = 16i+hb..16i+hb+7 -> 2 x b128
    float* o = ob + (size_t)t * BS_ * NST;
#pragma unroll
    for (int i = 0; i < 4; ++i) {
      *(v4f*)(o + 16 * i + hb)     =
          __builtin_shufflevector(c[i], c[i], 0, 1, 2, 3);
      *(v4f*)(o + 16 * i + hb + 4) =
          __builtin_shufflevector(c[i], c[i], 4, 5, 6, 7);
    }
  }
}

extern "C" void kernel_launch(void* const* d_in, const int* in_sizes, int n_in,
                              void* d_out, int out_size, void* d_ws,
                              size_t ws_size, hipStream_t stream) {
  (void)in_sizes; (void)n_in; (void)out_size; (void)ws_size;
  const float* X  = (const float*)d_in[0];   // (L, BS)
  const float* dA = (const float*)d_in[1];   // (64, 64)
  const float* dB = (const float*)d_in[2];   // (64)
  float* OUT = (float*)d_out;                // (L, BS, 64)
  float* P = (float*)d_ws;                   // 4096 floats: dA^64
  float* S = P + 4096;                       // GCH*BT*1024 floats = 8 MB

  hippo_prep<<<1, 128, 0, stream>>>(dA, P);
  hippo_chunk_end<<<GCH * BT, 32, 0, stream>>>(X, dA, dB, S);
  hippo_scan<<<1, 512, 0, stream>>>(P, S);
  hippo_emit<<<GCH * BT, 32, 0, stream>>>(X, dA, dB, S, OUT);
}